// CRF_63118839382371
// MI455X (gfx1250) — compile-verified
//
#include <hip/hip_runtime.h>
#include <hip/hip_bf16.h>
#include <math.h>

#define BB 512
#define SS 1024
#define TT 48
#define PAD_TAG 0
#define START_TAG 1
#define END_TAG 2
#define ESTRIDE 52   // floats; 8B-aligned rows, conflict-free bank spread

typedef __attribute__((ext_vector_type(2))) float v2f;
typedef __attribute__((ext_vector_type(8))) float v8f;

// ---------------------------------------------------------------------------
// Kernel 1: true path score per batch (gathers + masked sum)
// ---------------------------------------------------------------------------
__global__ __launch_bounds__(256) void crf_true_score(
    const float* __restrict__ P, const float* __restrict__ A,
    const int* __restrict__ y, const int* __restrict__ mask,
    float* __restrict__ ts)
{
    __shared__ float red[256];
    __shared__ int   redc[256];
    const int b = blockIdx.x, tid = threadIdx.x;
    float acc = 0.f; int cnt = 0;
    for (int s = tid; s < SS; s += 256) {
        int ys = y[b * SS + s];
        int ms = mask[b * SS + s];
        cnt += (ms > 0) ? 1 : 0;
        if (s >= 1 && ms > 0) {
            int yp = y[b * SS + s - 1];
            acc += A[yp * TT + ys] + P[((size_t)b * SS + s) * TT + ys];
        }
    }
    red[tid] = acc; redc[tid] = cnt;
    __syncthreads();
    for (int off = 128; off > 0; off >>= 1) {
        if (tid < off) { red[tid] += red[tid + off]; redc[tid] += redc[tid + off]; }
        __syncthreads();
    }
    if (tid == 0) {
        int   y0 = y[b * SS];
        float sc = red[0] + A[START_TAG * TT + y0] + P[(size_t)b * SS * TT + y0];
        int   lastIdx = redc[0] - 1;
        if (lastIdx < 0) lastIdx = 0;
        int   lt = y[b * SS + lastIdx];
        sc += A[lt * TT + END_TAG];
        ts[b] = sc;
    }
}

// ---------------------------------------------------------------------------
// Kernel 2: forward scan (logZ) via WMMA.
// One wave handles 16 batches. Step: m=rowmax(alpha); E=exp(alpha-m);
// Sacc = E(16x48) x M(48x48) with 36 x V_WMMA_F32_16X16X4_F32;
// alpha' = log(Sacc) + m + P[t].
// ---------------------------------------------------------------------------
__global__ __launch_bounds__(32) void crf_forward_wmma(
    const float* __restrict__ P, const float* __restrict__ A,
    const int* __restrict__ mask, float* __restrict__ logZ)
{
    __shared__ float aL[16 * ESTRIDE];  // alpha[b][j]
    __shared__ float eL[16 * ESTRIDE];  // exp(alpha-m)[b][i]
    __shared__ float mL[16];            // per-row max

    const int lane = threadIdx.x;
    const int lo = lane & 15, hi = lane >> 4;
    const int b0 = blockIdx.x * 16;

    // Hoisted B-fragments of M = exp(A) (exact 0/1). [K][N] layout:
    // lane holds M[kc*4 + 2*hi + {0,1}][jt*16 + lo].
    v2f bfrag[3][12];
#pragma unroll
    for (int jt = 0; jt < 3; ++jt)
#pragma unroll
        for (int kc = 0; kc < 12; ++kc) {
            int k = kc * 4 + 2 * hi;
            int n = jt * 16 + lo;
            bfrag[jt][kc].x = __expf(A[k * TT + n]);
            bfrag[jt][kc].y = __expf(A[(k + 1) * TT + n]);
        }

    // alpha0[b][j] = A[START][j] + P[b][0][j]; lane covers b=lo, 24 i's per half
    {
        const float* Pb = P + (size_t)(b0 + lo) * SS * TT;
#pragma unroll
        for (int q = 0; q < 24; ++q) {
            int i = hi * 24 + q;
            aL[lo * ESTRIDE + i] = A[START_TAG * TT + i] + Pb[i];
        }
    }
    __syncthreads();

    for (int t = 1; t < SS; ++t) {
        // --- phase 1: row max + exponentials ---
        float mloc = -INFINITY;
#pragma unroll
        for (int q = 0; q < 24; ++q)
            mloc = fmaxf(mloc, aL[lo * ESTRIDE + hi * 24 + q]);
        float mrow = fmaxf(mloc, __shfl_xor(mloc, 16, 32));
#pragma unroll
        for (int q = 0; q < 24; ++q) {
            int i = hi * 24 + q;
            eL[lo * ESTRIDE + i] = __expf(aL[lo * ESTRIDE + i] - mrow);
        }
        if (hi == 0) mL[lo] = mrow;
        int mk = mask[(b0 + lo) * SS + t];
        __syncthreads();

        // --- phase 2: 36 WMMAs (3 N-tiles x 12 K-chunks) ---
        v8f acc0 = {}, acc1 = {}, acc2 = {};
#pragma unroll
        for (int kc = 0; kc < 12; ++kc) {
            v2f af;
            const float* ep = &eL[lo * ESTRIDE + kc * 4 + 2 * hi];
            af.x = ep[0]; af.y = ep[1];
            acc0 = __builtin_amdgcn_wmma_f32_16x16x4_f32(false, af, false, bfrag[0][kc],
                                                         (short)0, acc0, false, false);
            acc1 = __builtin_amdgcn_wmma_f32_16x16x4_f32(false, af, false, bfrag[1][kc],
                                                         (short)0, acc1, false, false);
            acc2 = __builtin_amdgcn_wmma_f32_16x16x4_f32(false, af, false, bfrag[2][kc],
                                                         (short)0, acc2, false, false);
        }
        __syncthreads();  // everyone done reading eL/aL

        // --- phase 3: epilogue. D-layout: vgpr v -> b = v + 8*hi, n = jt*16+lo
#pragma unroll
        for (int v = 0; v < 8; ++v) {
            int bb = v + 8 * hi;
            float mb = mL[bb];
            int   mv = __shfl(mk, bb, 32);
            const float* Pr = P + ((size_t)(b0 + bb) * SS + t) * TT;
            float nj0 = __logf(acc0[v]) + mb + Pr[0 * 16 + lo];
            float nj1 = __logf(acc1[v]) + mb + Pr[1 * 16 + lo];
            float nj2 = __logf(acc2[v]) + mb + Pr[2 * 16 + lo];
            float* ar = &aL[bb * ESTRIDE];
            if (mv > 0) {
                ar[0 * 16 + lo] = nj0;
                ar[1 * 16 + lo] = nj1;
                ar[2 * 16 + lo] = nj2;
            }
            if (t + 1 < SS) __builtin_prefetch(Pr + TT, 0, 0);  // global_prefetch_b8
        }
        __syncthreads();
    }

    // logZ[b] = logsumexp_j(alpha[b][j] + A[j][END])
    {
        float vbuf[24];
        float mloc = -INFINITY;
#pragma unroll
        for (int q = 0; q < 24; ++q) {
            int j = hi * 24 + q;
            vbuf[q] = aL[lo * ESTRIDE + j] + A[j * TT + END_TAG];
            mloc = fmaxf(mloc, vbuf[q]);
        }
        float mrow = fmaxf(mloc, __shfl_xor(mloc, 16, 32));
        float ssum = 0.f;
#pragma unroll
        for (int q = 0; q < 24; ++q) ssum += __expf(vbuf[q] - mrow);
        ssum += __shfl_xor(ssum, 16, 32);
        if (hi == 0) logZ[b0 + lo] = mrow + __logf(ssum);
    }
}

// ---------------------------------------------------------------------------
// Kernel 3: Viterbi (max-plus scan) — backpointers live entirely in LDS.
// ---------------------------------------------------------------------------
__global__ __launch_bounds__(64) void crf_viterbi(
    const float* __restrict__ P, const float* __restrict__ A,
    const int* __restrict__ mask, float* __restrict__ pred)
{
    __shared__ float sA[TT * TT];
    __shared__ float al[TT];
    __shared__ unsigned char bp[(SS - 1) * TT];   // ~49 KB, fits easily in 320 KB LDS

    const int j = threadIdx.x;
    const int b = blockIdx.x;
    for (int x = j; x < TT * TT; x += 64) sA[x] = A[x];
    if (j < TT) al[j] = A[START_TAG * TT + j] + P[(size_t)b * SS * TT + j];
    __syncthreads();

    for (int t = 1; t < SS; ++t) {
        float best = -INFINITY; int arg = 0;
        if (j < TT) {
            for (int i = 0; i < TT; ++i) {
                float sc = al[i] + sA[i * TT + j];
                if (sc > best) { best = sc; arg = i; }   // first-max, matches argmax
            }
        }
        __syncthreads();
        if (j < TT) {
            bp[(t - 1) * TT + j] = (unsigned char)arg;
            if (mask[b * SS + t] > 0)
                al[j] = best + P[((size_t)b * SS + t) * TT + j];
            if (t + 2 < SS)
                __builtin_prefetch(&P[((size_t)b * SS + t + 2) * TT + j], 0, 0);
        }
        __syncthreads();
    }

    if (j == 0) {
        float best = -INFINITY; int arg = 0;
        for (int i = 0; i < TT; ++i) {
            float sc = al[i] + sA[i * TT + END_TAG];
            if (sc > best) { best = sc; arg = i; }
        }
        int tcur = arg;
        for (int s = SS - 1; s >= 1; --s) {
            pred[(size_t)b * SS + s] =
                (float)((mask[b * SS + s] > 0) ? tcur : PAD_TAG);
            tcur = (int)bp[(s - 1) * TT + tcur];
        }
        pred[(size_t)b * SS + 0] = (float)tcur;
    }
}

// ---------------------------------------------------------------------------
// Kernel 4: deterministic loss reduction: loss = sum_b (logZ[b] - true[b])
// ---------------------------------------------------------------------------
__global__ __launch_bounds__(256) void crf_loss_reduce(
    const float* __restrict__ logZ, const float* __restrict__ ts,
    float* __restrict__ out)
{
    __shared__ float red[256];
    const int tid = threadIdx.x;
    float a = 0.f;
    for (int b = tid; b < BB; b += 256) a += logZ[b] - ts[b];
    red[tid] = a;
    __syncthreads();
    for (int off = 128; off > 0; off >>= 1) {
        if (tid < off) red[tid] += red[tid + off];
        __syncthreads();
    }
    if (tid == 0) out[0] = red[0];
}

// ---------------------------------------------------------------------------
extern "C" void kernel_launch(void* const* d_in, const int* in_sizes, int n_in,
                              void* d_out, int out_size, void* d_ws, size_t ws_size,
                              hipStream_t stream)
{
    (void)in_sizes; (void)n_in; (void)out_size; (void)ws_size;
    const float* P    = (const float*)d_in[0];
    const float* A    = (const float*)d_in[1];
    const int*   y    = (const int*)d_in[2];
    const int*   mask = (const int*)d_in[3];
    float* out = (float*)d_out;          // [0]=loss, [1..] = pred (B*S)
    float* ws_logZ = (float*)d_ws;       // 512 floats
    float* ws_ts   = ws_logZ + BB;       // 512 floats

    crf_true_score<<<BB, 256, 0, stream>>>(P, A, y, mask, ws_ts);
    crf_forward_wmma<<<BB / 16, 32, 0, stream>>>(P, A, mask, ws_logZ);
    crf_viterbi<<<BB, 64, 0, stream>>>(P, A, mask, out + 1);
    crf_loss_reduce<<<1, 256, 0, stream>>>(ws_logZ, ws_ts, out);
}